// DeepSeekMoE_6777458393401
// MI455X (gfx1250) — compile-verified
//
#include <hip/hip_runtime.h>
#include <hip/hip_bf16.h>

typedef __attribute__((ext_vector_type(2))) float v2f;
typedef __attribute__((ext_vector_type(4))) float v4f;
typedef __attribute__((ext_vector_type(8))) float v8f;

#define H_DIM   4096
#define IN_DIM  512
#define E_DIM   8
#define OUT_DIM 512
#define NTOK    16384   // B*S = 8*2048

// ---------------------------------------------------------------------------
// Prep 1: Weff[e][k] = sum_h Wr[e][h]*Wi[h][k]  (router folded through in_proj)
//         beff[e]    = sum_h Wr[e][h]*bi[h] + br[e]
// Writes Weff into rows 0..7 of WcT (combined weight, [16][512] n-major).
// grid 16 x 256 : one thread per (e,k).
// ---------------------------------------------------------------------------
__global__ void moe_prep_weff(const float* __restrict__ Wi, const float* __restrict__ Wr,
                              const float* __restrict__ bi, const float* __restrict__ br,
                              float* __restrict__ WcT, float* __restrict__ beff) {
    int t = blockIdx.x * blockDim.x + threadIdx.x;   // 0..4095
    int e = t >> 9;                                  // 0..7
    int k = t & 511;                                 // 0..511
    const float* wr = Wr + (size_t)e * H_DIM;
    float acc = 0.f;
    for (int h = 0; h < H_DIM; ++h)
        acc = __builtin_fmaf(wr[h], Wi[(size_t)h * IN_DIM + k], acc);
    WcT[e * IN_DIM + k] = acc;
    if (k == 0) {
        float b = br[e];
        for (int h = 0; h < H_DIM; ++h)
            b = __builtin_fmaf(wr[h], bi[h], b);
        beff[e] = b;
    }
}

// ---------------------------------------------------------------------------
// Prep 2: rows 8..15 of WcT = Wi[0:8][:] (the gathered hidden channels),
//         wsum[o] = sum_h Wo[o][h]  (out_proj collapsed against broadcast agg)
// grid 2 x 256 : one thread per k / per o.
// ---------------------------------------------------------------------------
__global__ void moe_prep_misc(const float* __restrict__ Wi, const float* __restrict__ Wo,
                              float* __restrict__ WcT, float* __restrict__ wsum) {
    int k = blockIdx.x * blockDim.x + threadIdx.x;   // 0..511
    #pragma unroll
    for (int n = 0; n < 8; ++n)
        WcT[(8 + n) * IN_DIM + k] = Wi[(size_t)n * IN_DIM + k];
    const float* wo = Wo + (size_t)k * H_DIM;
    float s = 0.f;
    for (int h = 0; h < H_DIM; ++h) s += wo[h];
    wsum[k] = s;
}

// ---------------------------------------------------------------------------
// Main kernel: one wave32 per 16-token tile (1024 tiles).
//   C[16 tok][16 cols] = X_tile[16x512] @ WcT^T[512x16]  via 128x
//   V_WMMA_F32_16X16X4_F32. Cols 0..7 = router logits, cols 8..15 = h[...,0:8].
//   Epilogue: top-2 argmax of logits (softmax is monotonic), agg scalar,
//   rank-1 out = agg*wsum + bo streamed with b128 stores.
// ---------------------------------------------------------------------------
__global__ void __launch_bounds__(256)
moe_main(const float* __restrict__ x,    const float* __restrict__ WcT,
         const float* __restrict__ beff, const float* __restrict__ bi,
         const float* __restrict__ wsum, const float* __restrict__ bo,
         float* __restrict__ out) {
    __shared__ float lds_c[8][16][16];   // per-wave 16x16 result tile
    __shared__ float lds_agg[8][16];     // per-wave agg scalar per token

    const int wave  = threadIdx.x >> 5;
    const int lane  = threadIdx.x & 31;
    const int tile  = blockIdx.x * 8 + wave;
    const int tok0  = tile * 16;
    const int col   = lane & 15;   // A: row M ; B: column N ; C: column N
    const int half  = lane >> 4;   // K sub-offset select (ISA 16x4 f32 layout)

    // A lane L<16: (M=L, K=k..k+1); lane L+16: (M=L, K=k+2..k+3) -> b64 loads
    const float* ap = x   + (size_t)(tok0 + col) * IN_DIM + half * 2;
    // B stored n-major [16][512] so it loads with the identical pattern
    const float* bp = WcT + (size_t)col * IN_DIM + half * 2;

    v8f acc = {};
    #pragma unroll 8
    for (int k = 0; k < IN_DIM; k += 4) {
        v2f a = *(const v2f*)(ap + k);
        v2f b = *(const v2f*)(bp + k);
        acc = __builtin_amdgcn_wmma_f32_16x16x4_f32(
            /*neg_a=*/false, a, /*neg_b=*/false, b,
            /*c_mod=*/(short)0, acc, /*reuse_a=*/false, /*reuse_b=*/false);
    }

    // Preload rank-1 epilogue vectors (each lane owns 4x float4 of the 512 outs)
    v4f wv[4], bv[4];
    #pragma unroll
    for (int i = 0; i < 4; ++i) {
        wv[i] = *(const v4f*)(wsum + lane * 4 + 128 * i);
        bv[i] = *(const v4f*)(bo   + lane * 4 + 128 * i);
    }

    // Column bias: logits get beff, h-channels get bi[0:8]. Each lane holds one
    // column (N=col) across 8 rows (M = half*8 + j) -> transpose through LDS.
    float cb = (col < 8) ? beff[col] : bi[col - 8];
    #pragma unroll
    for (int j = 0; j < 8; ++j)
        lds_c[wave][half * 8 + j][col] = acc[j] + cb;
    __syncthreads();

    if (lane < 16) {
        const float* r = lds_c[wave][lane];      // one token's 16 values
        // top-2 of logits (first-index tie-break like lax.top_k)
        float b1 = r[0]; int i1 = 0;
        #pragma unroll
        for (int n = 1; n < 8; ++n) if (r[n] > b1) { b1 = r[n]; i1 = n; }
        float b2 = -3.402823466e38f; int i2 = 0;
        #pragma unroll
        for (int n = 0; n < 8; ++n) if (n != i1 && r[n] > b2) { b2 = r[n]; i2 = n; }
        // gathered = h[..., idx]; mean of top-2
        lds_agg[wave][lane] = 0.5f * (r[8 + i1] + r[8 + i2]);
    }
    __syncthreads();

    // out[t][o] = agg[t]*wsum[o] + bo[o] ; contiguous b128 stores per lane
    #pragma unroll 4
    for (int t = 0; t < 16; ++t) {
        float ag = lds_agg[wave][t];
        float* op = out + (size_t)(tok0 + t) * OUT_DIM + lane * 4;
        #pragma unroll
        for (int i = 0; i < 4; ++i) {
            v4f r = ag * wv[i] + bv[i];
            *(v4f*)(op + 128 * i) = r;
        }
    }
}

extern "C" void kernel_launch(void* const* d_in, const int* in_sizes, int n_in,
                              void* d_out, int out_size, void* d_ws, size_t ws_size,
                              hipStream_t stream) {
    const float* x  = (const float*)d_in[0];
    const float* Wi = (const float*)d_in[1];
    const float* bi = (const float*)d_in[2];
    const float* Wr = (const float*)d_in[3];
    const float* br = (const float*)d_in[4];
    const float* Wo = (const float*)d_in[5];
    const float* bo = (const float*)d_in[6];
    float* out = (float*)d_out;

    // Workspace layout (floats): WcT[16*512] | beff[8]+pad | wsum[512]  (~35 KB)
    float* WcT  = (float*)d_ws;
    float* beff = WcT + 16 * IN_DIM;
    float* wsum = beff + 16;

    moe_prep_weff<<<16, 256, 0, stream>>>(Wi, Wr, bi, br, WcT, beff);
    moe_prep_misc<<<2, 256, 0, stream>>>(Wi, Wo, WcT, wsum);
    moe_main<<<NTOK / 16 / 8, 256, 0, stream>>>(x, WcT, beff, bi, wsum, bo, out);
}